// HoltWintersDecomposition_40192303956464
// MI455X (gfx1250) — compile-verified
//
#include <hip/hip_runtime.h>
#include <math.h>

// Holt-Winters decomposition scan for MI455X (gfx1250, wave32).
// x: (64, 4096, 128) f32. Outputs: level, season, y concatenated flat.
//
// Structure: 256 single-wave workgroups; each wave owns one (batch, 32-feature)
// slice and walks t serially. Async global->LDS double buffering (ASYNCcnt)
// hides DRAM latency ahead of the serial dependent chain; NT stores stream the
// 402MB of output without polluting L2.

#define EPS 1e-8f

constexpr int B = 64;
constexpr int T = 4096;
constexpr int D = 128;
constexpr int CHUNK = 64;              // timesteps per prefetch chunk
constexpr int NCH = T / CHUNK;         // 64 chunks
constexpr long long NTOT = (long long)B * T * D;   // 33554432 elements per output

// Fast reciprocal: v_rcp_f32 (TRANS pipe) + one Newton-Raphson step (~0.5 ulp).
__device__ __forceinline__ float rcp_nr(float v) {
    float r = __builtin_amdgcn_rcpf(v);
    float e = __builtin_fmaf(-v, r, 1.0f);
    return __builtin_fmaf(r, e, r);
}

// Issue one chunk (CHUNK rows x 32 floats = 8KB) of async global->LDS copies.
// Per issue: 32 lanes x 16B = 512B, so 16 issues per chunk. ASYNCcnt += 16.
__device__ __forceinline__ void async_load_chunk(const float* sbase,
                                                 unsigned goff,
                                                 unsigned ldsoff) {
#pragma unroll
    for (int i = 0; i < 16; ++i) {
        unsigned g = goff + (unsigned)i * 2048u;   // 4 rows * 512B per issue
        unsigned l = ldsoff + (unsigned)i * 512u;  // 4 rows * 128B per issue
        asm volatile("global_load_async_to_lds_b128 %0, %1, %2"
                     :
                     : "v"(l), "v"(g), "s"(sbase)
                     : "memory");
    }
}

__global__ void HoltWintersDecomposition_kernel(const float* __restrict__ x,
                                                const float* __restrict__ p_la,
                                                const float* __restrict__ p_lg,
                                                float* __restrict__ out) {
    const int lane = threadIdx.x;          // 0..31 (wave32)
    const int b    = blockIdx.x >> 2;      // batch
    const int g    = blockIdx.x & 3;       // 32-wide feature group
    const int d    = g * 32 + lane;        // feature index 0..127

    __shared__ float tile[2][CHUNK][32];   // 16KB double buffer, wave-private

    // Scalars (single-element device arrays); sigmoid on device.
    const float la    = p_la[0];
    const float lg    = p_lg[0];
    const float alpha = 1.0f / (1.0f + __expf(-la));
    const float gamma = 1.0f / (1.0f + __expf(-lg));
    const float oma   = 1.0f - alpha;
    const float omg   = 1.0f - gamma;

    const float* xb = x + (long long)b * T * D;
    float* outL = out;
    float* outS = out + NTOT;
    float* outY = out + 2 * NTOT;
    const long long obase = (long long)b * T * D + d;

    // Per-lane address pattern inside one async issue:
    // 8 lanes cover one 128B row-slice (4 floats each x 8 = 32 floats),
    // 4 rows per issue.
    const unsigned rowInIssue = (unsigned)lane >> 3;
    const unsigned seg        = (unsigned)lane & 7u;
    const unsigned laneG = rowInIssue * 512u + (unsigned)g * 128u + seg * 16u;
    const unsigned laneL = rowInIssue * 128u + seg * 16u;
    // Low 32 bits of a generic LDS pointer are the wave-relative LDS byte addr.
    const unsigned ldsBase = (unsigned)(uintptr_t)(&tile[0][0][0]);

    // Prologue: prefetch chunks 0 and 1 (32 outstanding async ops).
    async_load_chunk(xb, 0u * 32768u + laneG, ldsBase + 0u * 8192u + laneL);
    async_load_chunk(xb, 1u * 32768u + laneG, ldsBase + 1u * 8192u + laneL);

    // t = 0: level = x0, season = 1, y0 = x0/(x0+eps).
    float x0     = xb[d];
    float level  = x0;
    float season = 1.0f;
    {
        float y0 = x0 * rcp_nr(__builtin_fmaf(level, season, EPS));
        __builtin_nontemporal_store(level,  outL + obase);
        __builtin_nontemporal_store(season, outS + obase);
        __builtin_nontemporal_store(y0,     outY + obase);
    }

    for (int c = 0; c < NCH; ++c) {
        // Async loads complete in order: <=16 outstanding means chunk c landed
        // while chunk c+1 may still be in flight. Last chunk: drain fully.
        if (c < NCH - 1) {
            asm volatile("s_wait_asynccnt 0x10" ::: "memory");
        } else {
            asm volatile("s_wait_asynccnt 0x0" ::: "memory");
        }

        const int buf = c & 1;
        const int r0  = (c == 0) ? 1 : 0;   // chunk 0 row 0 already handled
        long long o = obase + (long long)(c * CHUNK + r0) * D;

#pragma unroll 4
        for (int r = r0; r < CHUNK; ++r) {
            float xt = tile[buf][r][lane];

            float inv_s = rcp_nr(season + EPS);
            level  = __builtin_fmaf(alpha, xt * inv_s, oma * level);
            float inv_l = rcp_nr(level + EPS);
            season = __builtin_fmaf(gamma, xt * inv_l, omg * season);
            float y = xt * rcp_nr(__builtin_fmaf(level, season, EPS));

            __builtin_nontemporal_store(level,  outL + o);
            __builtin_nontemporal_store(season, outS + o);
            __builtin_nontemporal_store(y,      outY + o);
            o += D;
        }

        // Refill the buffer we just drained with chunk c+2.
        if (c + 2 < NCH) {
            asm volatile("s_wait_dscnt 0x0" ::: "memory"); // LDS reads retired
            async_load_chunk(xb, (unsigned)(c + 2) * 32768u + laneG,
                             ldsBase + (unsigned)buf * 8192u + laneL);
        }
    }
}

extern "C" void kernel_launch(void* const* d_in, const int* in_sizes, int n_in,
                              void* d_out, int out_size, void* d_ws, size_t ws_size,
                              hipStream_t stream) {
    const float* x    = (const float*)d_in[0];
    const float* p_la = (const float*)d_in[1];
    const float* p_lg = (const float*)d_in[2];
    float* out        = (float*)d_out;

    // 256 single-wave (32-thread) workgroups: 64 batches x 4 feature groups.
    dim3 grid(B * 4);
    dim3 block(32);
    HoltWintersDecomposition_kernel<<<grid, block, 0, stream>>>(x, p_la, p_lg, out);
}